// GAT_40733469835582
// MI455X (gfx1250) — compile-verified
//
#include <hip/hip_runtime.h>
#include <hip/hip_bf16.h>
#include <math.h>

// ---------------------------------------------------------------------------
// GAT (3-layer, PyG-style) for MI455X / gfx1250, wave32.
// Dense GEMMs use V_WMMA_F32_16X16X4_F32 (fp32 matrix pipe, matches reference
// precision). Edge softmax/aggregation via L2-resident atomics (h fits in the
// 192MB L2: 50000*128*4B = 25.6MB).
// ---------------------------------------------------------------------------

typedef __attribute__((ext_vector_type(2))) float v2f;
typedef __attribute__((ext_vector_type(8))) float v8f;

#define NEG_SLOPE 0.2f
#define BN_EPS 1e-5f
#define HHEADS 4

// Monotone float <-> uint mapping for atomic float-max via atomicMax(u32).
__device__ __forceinline__ unsigned f2ord(float f) {
  unsigned u = __float_as_uint(f);
  return (u & 0x80000000u) ? ~u : (u | 0x80000000u);
}
__device__ __forceinline__ float ord2f(unsigned u) {
  unsigned b = (u & 0x80000000u) ? (u & 0x7FFFFFFFu) : ~u;
  return __uint_as_float(b);
}
// f2ord(-inf) = ~0xFF800000 = 0x007FFFFF
#define ORD_NEG_INF 0x007FFFFFu

__global__ void fill_u32(unsigned* __restrict__ p, unsigned v, int n) {
  int i = blockIdx.x * blockDim.x + threadIdx.x;
  if (i < n) p[i] = v;
}

// ---------------------------------------------------------------------------
// C[M,Nout] = A[M,K] * B[K,Nout], fp32 WMMA.  One wave -> one 16x16 tile.
// A 16x4 layout: lanes 0-15 = rows M; lane-half selects K={0,1} vs {2,3}.
// B 4x16 layout: lanes 0-15 = cols N; lane-half selects K rows {0,1}/{2,3}.
// C/D: VGPR j -> row j (lanes 0-15) / row j+8 (lanes 16-31), col = lane&15.
// ---------------------------------------------------------------------------
__global__ void gemm_wmma_f32(const float* __restrict__ A,
                              const float* __restrict__ B,
                              float* __restrict__ C,
                              int M, int K, int Nout) {
  const int wave = threadIdx.x >> 5;
  const int lane = threadIdx.x & 31;
  const int half = lane >> 4;
  const int l    = lane & 15;

  const int mtile = blockIdx.x * (blockDim.x >> 5) + wave;
  const int ntile = blockIdx.y;
  const int m0 = mtile * 16;
  const int n0 = ntile * 16;
  if (m0 >= M) return;  // wave-uniform

  int arow = m0 + l;
  if (arow >= M) arow = M - 1;  // clamp (M multiple of 16 in practice)
  const float* __restrict__ ap = A + (size_t)arow * K + 2 * half;

  v8f acc = {};
  for (int k = 0; k < K; k += 4) {
    v2f a, b;
    a.x = ap[k + 0];
    a.y = ap[k + 1];
    b.x = B[(size_t)(k + 2 * half + 0) * Nout + n0 + l];
    b.y = B[(size_t)(k + 2 * half + 1) * Nout + n0 + l];
    acc = __builtin_amdgcn_wmma_f32_16x16x4_f32(
        /*neg_a=*/false, a, /*neg_b=*/false, b,
        /*c_mod=*/(short)0, acc, /*reuse_a=*/false, /*reuse_b=*/false);
  }

#pragma unroll
  for (int j = 0; j < 8; ++j) {
    int row = m0 + j + 8 * half;
    if (row < M) C[(size_t)row * Nout + n0 + l] = acc[j];
  }
}

// ---------------------------------------------------------------------------
// Attention logits: al_s/al_d[n,h] = dot(h[n,h,:], a_src/a_dst[h,:])
// ---------------------------------------------------------------------------
__global__ void attn_logits(const float* __restrict__ h,
                            const float* __restrict__ a_src,
                            const float* __restrict__ a_dst,
                            float* __restrict__ als,
                            float* __restrict__ ald,
                            int N, int C) {
  int i = blockIdx.x * blockDim.x + threadIdx.x;  // over N*H
  if (i >= N * HHEADS) return;
  int hd = i & (HHEADS - 1);
  const float* __restrict__ hp = h + (size_t)i * C;
  const float* __restrict__ as = a_src + hd * C;
  const float* __restrict__ ad = a_dst + hd * C;
  float s = 0.f, d = 0.f;
  for (int c = 0; c < C; ++c) {
    float v = hp[c];
    s += v * as[c];
    d += v * ad[c];
  }
  als[i] = s;
  ald[i] = d;
}

__device__ __forceinline__ void edge_ends(const int* __restrict__ ei, int E,
                                          int e, int& s, int& d) {
  if (e < E) { s = ei[e]; d = ei[E + e]; }
  else       { s = e - E; d = s; }      // self-loop
}

__device__ __forceinline__ float lrelu(float v) {
  return v > 0.f ? v : NEG_SLOPE * v;
}

// Pass 1: segment max of leaky_relu(al_s[src]+al_d[dst]) into ordered-u32.
__global__ void edge_max(const int* __restrict__ ei, int E, int Etot,
                         const float* __restrict__ als,
                         const float* __restrict__ ald,
                         unsigned* __restrict__ mord) {
  int t = blockIdx.x * blockDim.x + threadIdx.x;
  if (t >= Etot * HHEADS) return;
  int e = t >> 2, hd = t & 3;
  int s, d;
  edge_ends(ei, E, e, s, d);
  float v = lrelu(als[s * HHEADS + hd] + ald[d * HHEADS + hd]);
  atomicMax(&mord[d * HHEADS + hd], f2ord(v));
}

// Decode ordered-u32 max back to float (in place).
__global__ void decode_max(unsigned* __restrict__ m, int n) {
  int i = blockIdx.x * blockDim.x + threadIdx.x;
  if (i < n) m[i] = __float_as_uint(ord2f(m[i]));
}

// Pass 2: denom[dst,h] += exp(e - max[dst,h])
__global__ void edge_denom(const int* __restrict__ ei, int E, int Etot,
                           const float* __restrict__ als,
                           const float* __restrict__ ald,
                           const float* __restrict__ mmax,
                           float* __restrict__ denom) {
  int t = blockIdx.x * blockDim.x + threadIdx.x;
  if (t >= Etot * HHEADS) return;
  int e = t >> 2, hd = t & 3;
  int s, d;
  edge_ends(ei, E, e, s, d);
  int di = d * HHEADS + hd;
  float v  = lrelu(als[s * HHEADS + hd] + ald[di]);
  float ex = __expf(v - mmax[di]);
  atomicAdd(&denom[di], ex);
}

// Pass 3: out[dst,h,:] += alpha * h[src,h,:]
__global__ void edge_aggr(const int* __restrict__ ei, int E, int Etot,
                          const float* __restrict__ als,
                          const float* __restrict__ ald,
                          const float* __restrict__ mmax,
                          const float* __restrict__ denom,
                          const float* __restrict__ h,
                          float* __restrict__ out, int C) {
  int t = blockIdx.x * blockDim.x + threadIdx.x;
  if (t >= Etot * HHEADS) return;
  int e = t >> 2, hd = t & 3;
  int s, d;
  edge_ends(ei, E, e, s, d);
  int di = d * HHEADS + hd;
  float v     = lrelu(als[s * HHEADS + hd] + ald[di]);
  float alpha = __expf(v - mmax[di]) / (denom[di] + 1e-16f);
  const float* __restrict__ hs = h + (size_t)(s * HHEADS + hd) * C;
  float* __restrict__ op = out + (size_t)di * C;
  for (int c = 0; c < C; ++c) atomicAdd(&op[c], hs[c] * alpha);
}

// ---------------------------------------------------------------------------
// BatchNorm column statistics (population variance, ddof=0) + fused ReLU.
// NOTE: additive bias before BN cancels exactly (shifts mean equally), so the
// GATConv biases b0/b1 are skipped with no loss of exactness.
// ---------------------------------------------------------------------------
__global__ void bn_stats(const float* __restrict__ in, int nrows, int ncols,
                         float* __restrict__ stats /* [2*ncols]: mu, rstd */) {
  int col = blockIdx.x;
  __shared__ float ss[256], sq[256];
  float s = 0.f, q = 0.f;
  for (int r = threadIdx.x; r < nrows; r += blockDim.x) {
    float v = in[(size_t)r * ncols + col];
    s += v;
    q += v * v;
  }
  ss[threadIdx.x] = s;
  sq[threadIdx.x] = q;
  __syncthreads();
  for (int o = 128; o > 0; o >>= 1) {
    if ((int)threadIdx.x < o) {
      ss[threadIdx.x] += ss[threadIdx.x + o];
      sq[threadIdx.x] += sq[threadIdx.x + o];
    }
    __syncthreads();
  }
  if (threadIdx.x == 0) {
    float inv = 1.f / (float)nrows;
    float mu  = ss[0] * inv;
    float var = sq[0] * inv - mu * mu;
    stats[col]         = mu;
    stats[ncols + col] = rsqrtf(var + BN_EPS);
  }
}

__global__ void bn_relu(const float* __restrict__ in,
                        const float* __restrict__ stats,
                        const float* __restrict__ g,
                        const float* __restrict__ beta,
                        float* __restrict__ outp, int nrows, int ncols) {
  int i = blockIdx.x * blockDim.x + threadIdx.x;
  if (i >= nrows * ncols) return;
  int c = i % ncols;
  float v = g[c] * (in[i] - stats[c]) * stats[ncols + c] + beta[c];
  outp[i] = v > 0.f ? v : 0.f;
}

// Final layer: mean over 4 heads + bias b2 -> [N, 40] fp32 output.
__global__ void head_mean_bias(const float* __restrict__ aggr,
                               const float* __restrict__ b2,
                               float* __restrict__ out, int N, int C) {
  int i = blockIdx.x * blockDim.x + threadIdx.x;
  if (i >= N * C) return;
  int n = i / C, c = i - n * C;
  float s = 0.f;
#pragma unroll
  for (int hd = 0; hd < HHEADS; ++hd)
    s += aggr[(size_t)(n * HHEADS + hd) * C + c];
  out[i] = 0.25f * s + b2[c];
}

// ---------------------------------------------------------------------------
extern "C" void kernel_launch(void* const* d_in, const int* in_sizes, int n_in,
                              void* d_out, int out_size, void* d_ws,
                              size_t ws_size, hipStream_t stream) {
  const float* x   = (const float*)d_in[0];
  const int*   ei  = (const int*)d_in[1];
  const float* W0  = (const float*)d_in[2];
  const float* as0 = (const float*)d_in[3];
  const float* ad0 = (const float*)d_in[4];
  const float* g0  = (const float*)d_in[6];
  const float* be0 = (const float*)d_in[7];
  const float* W1  = (const float*)d_in[8];
  const float* as1 = (const float*)d_in[9];
  const float* ad1 = (const float*)d_in[10];
  const float* g1  = (const float*)d_in[12];
  const float* be1 = (const float*)d_in[13];
  const float* W2  = (const float*)d_in[14];
  const float* as2 = (const float*)d_in[15];
  const float* ad2 = (const float*)d_in[16];
  const float* b2  = (const float*)d_in[17];

  const int N    = in_sizes[0] / 128;
  const int E    = in_sizes[1] / 2;
  const int Etot = E + N;

  // Workspace layout (floats). h fits in L2 -> edge phase is L2-resident.
  float* ws = (float*)d_ws;
  size_t off = 0;
  float* hbuf   = ws + off; off += (size_t)N * 160;
  float* outbuf = ws + off; off += (size_t)N * 160;
  float* xcur   = ws + off; off += (size_t)N * 128;
  float* als    = ws + off; off += (size_t)N * HHEADS;
  float* ald    = ws + off; off += (size_t)N * HHEADS;
  float* mbuf   = ws + off; off += (size_t)N * HHEADS;
  float* denom  = ws + off; off += (size_t)N * HHEADS;
  float* stats  = ws + off; off += 2 * 160;
  (void)ws_size; (void)n_in; (void)out_size;

  auto cdiv = [](long long a, long long b) { return (int)((a + b - 1) / b); };

  auto layer = [&](const float* xin, const float* W, const float* asrc,
                   const float* adst, int C) {
    const int Nout   = HHEADS * C;
    const int Mtiles = (N + 15) / 16;
    dim3 gg(cdiv(Mtiles, 8), Nout / 16);
    gemm_wmma_f32<<<gg, 256, 0, stream>>>(xin, W, hbuf, N, 128, Nout);

    attn_logits<<<cdiv((long long)N * HHEADS, 256), 256, 0, stream>>>(
        hbuf, asrc, adst, als, ald, N, C);

    fill_u32<<<cdiv((long long)N * HHEADS, 256), 256, 0, stream>>>(
        (unsigned*)mbuf, ORD_NEG_INF, N * HHEADS);
    fill_u32<<<cdiv((long long)N * HHEADS, 256), 256, 0, stream>>>(
        (unsigned*)denom, 0u, N * HHEADS);
    fill_u32<<<cdiv((long long)N * Nout, 256), 256, 0, stream>>>(
        (unsigned*)outbuf, 0u, N * Nout);

    const int ET = Etot * HHEADS;
    edge_max<<<cdiv(ET, 256), 256, 0, stream>>>(ei, E, Etot, als, ald,
                                                (unsigned*)mbuf);
    decode_max<<<cdiv((long long)N * HHEADS, 256), 256, 0, stream>>>(
        (unsigned*)mbuf, N * HHEADS);
    edge_denom<<<cdiv(ET, 256), 256, 0, stream>>>(ei, E, Etot, als, ald, mbuf,
                                                  denom);
    edge_aggr<<<cdiv(ET, 256), 256, 0, stream>>>(ei, E, Etot, als, ald, mbuf,
                                                 denom, hbuf, outbuf, C);
  };

  // Layer 0: 128 -> 4x32 (concat 128), BN+ReLU (bias b0 cancels in BN)
  layer(x, W0, as0, ad0, 32);
  bn_stats<<<128, 256, 0, stream>>>(outbuf, N, 128, stats);
  bn_relu<<<cdiv((long long)N * 128, 256), 256, 0, stream>>>(
      outbuf, stats, g0, be0, xcur, N, 128);

  // Layer 1: 128 -> 4x32 (concat 128), BN+ReLU
  layer(xcur, W1, as1, ad1, 32);
  bn_stats<<<128, 256, 0, stream>>>(outbuf, N, 128, stats);
  bn_relu<<<cdiv((long long)N * 128, 256), 256, 0, stream>>>(
      outbuf, stats, g1, be1, xcur, N, 128);

  // Layer 2: 128 -> 4x40, mean over heads + b2
  layer(xcur, W2, as2, ad2, 40);
  head_mean_bias<<<cdiv((long long)N * 40, 256), 256, 0, stream>>>(
      outbuf, b2, (float*)d_out, N, 40);
}